// BernoulliDensity_59442347377115
// MI455X (gfx1250) — compile-verified
//
#include <hip/hip_runtime.h>
#include <math.h>

typedef float v2f __attribute__((ext_vector_type(2)));
typedef float v8f __attribute__((ext_vector_type(8)));

#define HIDDEN 128
#define OUTF   64
#define WSTRIDE 132   // LDS row stride in floats: 132 mod 64 = 4 -> conflict-free b64 reads

// ---- init: zero accumulator (d_out), deg = 1 (self-loop) ----
__global__ void gcn_init(float* __restrict__ acc, float* __restrict__ deg, int n) {
    int idx = blockIdx.x * blockDim.x + threadIdx.x;
    if (idx < n * OUTF) acc[idx] = 0.0f;
    if (idx < n) deg[idx] = 1.0f;
}

// ---- degree: atomicAdd over dst ----
__global__ void gcn_degree(const long long* __restrict__ ei, float* __restrict__ deg, int E) {
    int idx = blockIdx.x * blockDim.x + threadIdx.x;
    if (idx < E) {
        int d = (int)ei[(size_t)E + idx];
        atomicAdd(&deg[d], 1.0f);
    }
}

// ---- dinv = rsqrt(deg), in place (deg >= 1 always) ----
__global__ void gcn_dinv(float* __restrict__ deg, int n) {
    int idx = blockIdx.x * blockDim.x + threadIdx.x;
    if (idx < n) deg[idx] = rsqrtf(deg[idx]);
}

// ---- h = x @ W via V_WMMA_F32_16X16X4_F32 ----
// Block = 128 threads (4 waves). Each wave computes rows [row0, row0+16) x all 64 cols.
__global__ void __launch_bounds__(128)
gcn_gemm_wmma(const float* __restrict__ x, const float* __restrict__ W,
              float* __restrict__ h, int n) {
    __shared__ float ldsWt[OUTF * WSTRIDE];   // W transposed: ldsWt[n][k]

    int tid = threadIdx.x;
    // cooperative transpose load of W[128][64] (32 KB) into LDS
    for (int i = tid; i < HIDDEN * OUTF; i += 128) {
        int k = i >> 6;          // row of W
        int nn = i & 63;         // col of W
        ldsWt[nn * WSTRIDE + k] = W[i];
    }
    __syncthreads();

    int wave = tid >> 5;
    int lane = tid & 31;
    int row0 = blockIdx.x * 64 + wave * 16;
    if (row0 >= n) return;               // wave-uniform exit (after barrier)

    int m  = lane & 15;                  // A: M index / B,D: N sub-index
    int kh = (lane >> 4) << 1;           // A/B: K sub-offset {0,2}

    int arow = row0 + m;                 // clamp tail rows (stores are guarded)
    if (arow >= n) arow = n - 1;
    const float* xr = x + (size_t)arow * HIDDEN;

    v8f c0 = {}, c1 = {}, c2 = {}, c3 = {};

    for (int k0 = 0; k0 < HIDDEN; k0 += 4) {
        // A fragment: lane holds x[arow][k0+kh .. k0+kh+1]  (ISA 16x4 f32 layout)
        v2f a;
        a.x = xr[k0 + kh];
        a.y = xr[k0 + kh + 1];

        const float* w0 = &ldsWt[( 0 + m) * WSTRIDE + k0 + kh];
        const float* w1 = &ldsWt[(16 + m) * WSTRIDE + k0 + kh];
        const float* w2 = &ldsWt[(32 + m) * WSTRIDE + k0 + kh];
        const float* w3 = &ldsWt[(48 + m) * WSTRIDE + k0 + kh];

        v2f b0; b0.x = w0[0]; b0.y = w0[1];
        v2f b1; b1.x = w1[0]; b1.y = w1[1];
        v2f b2; b2.x = w2[0]; b2.y = w2[1];
        v2f b3; b3.x = w3[0]; b3.y = w3[1];

        c0 = __builtin_amdgcn_wmma_f32_16x16x4_f32(false, a, false, b0, (short)0, c0, false, false);
        c1 = __builtin_amdgcn_wmma_f32_16x16x4_f32(false, a, false, b1, (short)0, c1, false, false);
        c2 = __builtin_amdgcn_wmma_f32_16x16x4_f32(false, a, false, b2, (short)0, c2, false, false);
        c3 = __builtin_amdgcn_wmma_f32_16x16x4_f32(false, a, false, b3, (short)0, c3, false, false);
    }

    // D layout: c[i] = D[M = i + 8*(lane>>4)][N = lane&15]
    int rbase = row0 + ((lane >> 4) << 3);
#pragma unroll
    for (int i = 0; i < 8; ++i) {
        int r = rbase + i;
        if (r < n) {
            float* hp = h + (size_t)r * OUTF + m;
            hp[0]  = c0[i];
            hp[16] = c1[i];
            hp[32] = c2[i];
            hp[48] = c3[i];
        }
    }
}

// ---- scatter: one wave (32 lanes) per edge, float2 per lane, 2 f32 atomics ----
__global__ void __launch_bounds__(256)
gcn_scatter(const float* __restrict__ h, const float* __restrict__ dinv,
            const long long* __restrict__ ei, float* __restrict__ acc, int E) {
    int e = blockIdx.x * 8 + (threadIdx.x >> 5);
    if (e >= E) return;
    int lane = threadIdx.x & 31;

    int s = (int)ei[e];
    int d = (int)ei[(size_t)E + e];
    float nrm = dinv[s] * dinv[d];      // wave-uniform broadcast loads

    const float2* hp = (const float2*)(h + (size_t)s * OUTF);
    float2 v = hp[lane];
    float* ap = acc + (size_t)d * OUTF + lane * 2;
    atomicAdd(ap,     v.x * nrm);
    atomicAdd(ap + 1, v.y * nrm);
}

// ---- finalize: out = sigmoid(acc + h*dinv^2 + b), in place over d_out ----
__global__ void gcn_finalize(float* __restrict__ out, const float* __restrict__ h,
                             const float* __restrict__ dinv, const float* __restrict__ b,
                             int n) {
    int idx = blockIdx.x * blockDim.x + threadIdx.x;
    if (idx < n * OUTF) {
        int node = idx >> 6;
        int t    = idx & 63;
        float di = dinv[node];
        float val = out[idx] + h[idx] * di * di + b[t];
        out[idx] = 1.0f / (1.0f + __expf(-val));
    }
}

extern "C" void kernel_launch(void* const* d_in, const int* in_sizes, int n_in,
                              void* d_out, int out_size, void* d_ws, size_t ws_size,
                              hipStream_t stream) {
    const float*     x  = (const float*)d_in[0];
    const long long* ei = (const long long*)d_in[1];   // int64 [2, E]: row0=src, row1=dst
    const float*     W  = (const float*)d_in[2];
    const float*     b  = (const float*)d_in[3];

    int n = in_sizes[0] / HIDDEN;    // 100000
    int E = in_sizes[1] / 2;         // 1600000

    float* out = (float*)d_out;                    // also the scatter accumulator
    float* h   = (float*)d_ws;                     // [n, 64]
    float* deg = h + (size_t)n * OUTF;             // [n], becomes dinv in place

    int totNO = n * OUTF;

    gcn_init    <<<(totNO + 255) / 256, 256, 0, stream>>>(out, deg, n);
    gcn_degree  <<<(E + 255) / 256,     256, 0, stream>>>(ei, deg, E);
    gcn_dinv    <<<(n + 255) / 256,     256, 0, stream>>>(deg, n);
    gcn_gemm_wmma<<<(n + 63) / 64,      128, 0, stream>>>(x, W, h, n);
    gcn_scatter <<<(E + 7) / 8,         256, 0, stream>>>(h, deg, ei, out, E);
    gcn_finalize<<<(totNO + 255) / 256, 256, 0, stream>>>(out, h, deg, b, n);
}